// algorithm_fp_torch_21827023798274
// MI455X (gfx1250) — compile-verified
//
#include <hip/hip_runtime.h>
#include <math.h>

typedef __attribute__((ext_vector_type(2))) float v2f;
typedef __attribute__((ext_vector_type(8))) float v8f;

#define NT 64
#define KK 8
#define OUTER 5
#define INNER 10
#define RHO_C 0.1f
#define PROP_P 0.1f
#define PROP_C 0.9f

// Cooperative 8x8 Gaussian elimination (no pivoting; M = mu*I + PSD is safe).
// M: 8x8 in LDS (destroyed). Bm: 8 x m in LDS (overwritten with solution).
// Lane i (<8) owns row i. All 32 lanes must reach the barriers.
__device__ __forceinline__ void solve8(float* M, float* Bm, int m, int lane) {
  for (int k = 0; k < 8; ++k) {
    if (lane > k && lane < 8) {
      float f = M[lane * 8 + k] / M[k * 8 + k];
      for (int j = k; j < 8; ++j) M[lane * 8 + j] -= f * M[k * 8 + j];
      for (int j = 0; j < m; ++j) Bm[lane * m + j] -= f * Bm[k * m + j];
    }
    __syncthreads();
  }
  for (int k = 7; k >= 0; --k) {
    if (lane == k) {
      float inv = 1.0f / M[k * 8 + k];
      for (int j = 0; j < m; ++j) Bm[k * m + j] *= inv;
    }
    __syncthreads();
    if (lane < k) {
      float f = M[lane * 8 + k];
      for (int j = 0; j < m; ++j) Bm[lane * m + j] -= f * Bm[k * m + j];
    }
    __syncthreads();
  }
}

__global__ __launch_bounds__(32) void rsma_sca_kernel(
    const float* __restrict__ Hg, const int* __restrict__ PtPtr,
    float* __restrict__ out, int B) {
  const int b = blockIdx.x;
  const int lane = threadIdx.x;
  const float Pt = (float)PtPtr[0];
  const float* __restrict__ Hb = Hg + (size_t)b * NT * KK;

  __builtin_prefetch(Hb, 0, 1);  // global_prefetch_b8

  __shared__ float sG0[64];          // G0 = H^T H
  __shared__ float sW[64], sWn[64];  // P = H*W
  __shared__ float sA[64];           // A = G0*W = H^T P
  __shared__ float sM[64], sB[64];   // solver scratch
  __shared__ float sb2[8];
  __shared__ float slam[8], sbk2[8], sbck2[8], sc1[8], sc2b[8], sres[8];
  __shared__ float ss1[8], ss2[8], sc2[8];
  __shared__ float swc[8], swcn[8], sHt[8], sE[8];
  __shared__ float sred[32];
  __shared__ float stmp[8];
  __shared__ float slagF[9], slagL[9];

  // ---------------- G0 = H^T H  (WMMA f32 16x16x4, K-chunks of 4) ----------
  //
  // A[m][k] = H[t0+k][m], B[k][n] = H[t0+k][n] -> per-lane A and B register
  // contents are identical (Gram product), so one v2f feeds both operands.
  // Lanes with m>=8 (rows 8..15 of A / cols 8..15 of B) only influence
  // D rows/cols 8..15 which we never read, so instead of zeroing them
  // (which costs divergent branches around every load) we just wrap the
  // column index (m & 7): every lane does an unconditional valid load.
  {
    const int mc = lane & 7;          // wrapped user column (no divergence)
    const int kb = (lane >> 4) << 1;  // this half-wave's K pair base (0 or 2)
#if __has_builtin(__builtin_amdgcn_wmma_f32_16x16x4_f32)
    v8f acc = {0.f, 0.f, 0.f, 0.f, 0.f, 0.f, 0.f, 0.f};
#pragma unroll
    for (int t0 = 0; t0 < NT; t0 += 4) {
      v2f ab;
      ab.x = Hb[(t0 + kb) * KK + mc];
      ab.y = Hb[(t0 + kb + 1) * KK + mc];
      acc = __builtin_amdgcn_wmma_f32_16x16x4_f32(false, ab, false, ab,
                                                  (short)0, acc, false, false);
    }
    // D layout: VGPR r, lanes 0..15 hold (M=r, N=lane) -> G0[r][lane]
    if (lane < KK) {
#pragma unroll
      for (int r = 0; r < KK; ++r) sG0[r * KK + lane] = acc[r];
    }
#else
    for (int e = lane; e < 64; e += 32) {
      int i = e >> 3, j = e & 7;
      float s = 0.f;
      for (int t = 0; t < NT; ++t) s += Hb[t * KK + i] * Hb[t * KK + j];
      sG0[e] = s;
    }
#endif
  }
  __syncthreads();

  // ---------------- init_lambda (strongest user -> smallest weight) --------
  {
    float n[8];
    float nsum = 0.f;
#pragma unroll
    for (int j = 0; j < 8; ++j) {
      n[j] = sqrtf(sG0[j * 8 + j]);
      nsum += n[j];
    }
    int rank = 0;
    if (lane < 8) {
#pragma unroll
      for (int i = 0; i < 8; ++i)
        rank += (n[i] > n[lane]) || (n[i] == n[lane] && i < lane);
      stmp[rank] = n[lane] / nsum;  // p sorted by descending strength
    }
    __syncthreads();
    if (lane < 8) {
      float v = stmp[7 - rank];  // strongest gets smallest p
      slam[lane] = v * v;
    }
    __syncthreads();
    float ls = 0.f;
#pragma unroll
    for (int j = 0; j < 8; ++j) ls += slam[j];
    __syncthreads();
    if (lane < 8) slam[lane] /= ls;
    // init_P:  W = diag(sqrt(Pp)/||h_j||)
    const float Pp = PROP_P * Pt / (float)KK;
    for (int e = lane; e < 64; e += 32) {
      int i = e >> 3, j = e & 7;
      sW[e] = (i == j) ? (sqrtf(Pp) / n[j]) : 0.f;
    }
    __syncthreads();
  }

  // ---------------- p_c init: dominant eigvec of G0 via power iteration ----
  {
    if (lane < 8) swc[lane] = 1.0f;
    __syncthreads();
    for (int it = 0; it < 64; ++it) {
      float y = 0.f;
      if (lane < 8) {
#pragma unroll
        for (int i = 0; i < 8; ++i) y += sG0[lane * 8 + i] * swc[i];
      }
      __syncthreads();
      if (lane < 8) stmp[lane] = y;
      __syncthreads();
      float nn = 0.f;
#pragma unroll
      for (int i = 0; i < 8; ++i) nn += stmp[i] * stmp[i];
      nn = rsqrtf(nn);
      __syncthreads();
      if (lane < 8) swc[lane] = y * nn;
      __syncthreads();
    }
    // p_c = sqrt(Pc) * Hv/||Hv||  ->  w_c = sqrt(Pc)/sqrt(v^T G0 v) * v
    float g = 0.f;
    if (lane < 8) {
#pragma unroll
      for (int i = 0; i < 8; ++i) g += sG0[lane * 8 + i] * swc[i];
    }
    __syncthreads();
    if (lane < 8) stmp[lane] = g * swc[lane];
    __syncthreads();
    float vg = 0.f;
#pragma unroll
    for (int i = 0; i < 8; ++i) vg += stmp[i];
    float sc = sqrtf(PROP_C * Pt) * rsqrtf(vg);
    __syncthreads();
    if (lane < 8) swc[lane] *= sc;
    __syncthreads();
  }

  float mu = 10.0f / Pt;

  // A = G0 * Wsrc  (all 32 lanes, 2 elements each)
  auto matmulA = [&](const float* Wsrc) {
    for (int e = lane; e < 64; e += 32) {
      int i = e >> 3, j = e & 7;
      float s = 0.f;
#pragma unroll
      for (int t = 0; t < 8; ++t) s += sG0[i * 8 + t] * Wsrc[t * 8 + j];
      sA[e] = s;
    }
    __syncthreads();
  };

  // upd_P via Woodbury: two 8x8 SPD solves instead of 64x64.
  auto updP = [&](float mu_) {
    if (lane < 8) {
      float l = slam[lane];
      ss1[lane] = sqrtf(sbk2[lane] + l * sbck2[lane]);
      ss2[lane] = sqrtf(l * sbck2[lane]);
      sc2[lane] = l * sc2b[lane];  // lam * sqrt(1+a_ck) * b_ck
    }
    __syncthreads();
    // M1 = mu*I + diag(s1) G0 diag(s1);  B1 = diag(s1) G0 diag(c1)
    for (int e = lane; e < 64; e += 32) {
      int i = e >> 3, j = e & 7;
      float g = sG0[e];
      sM[e] = ss1[i] * g * ss1[j] + ((i == j) ? mu_ : 0.f);
      sB[e] = ss1[i] * g * sc1[j];
    }
    __syncthreads();
    solve8(sM, sB, 8, lane);
    const float invmu = 1.0f / mu_;
    for (int e = lane; e < 64; e += 32) {
      int i = e >> 3, j = e & 7;
      sWn[e] = (((i == j) ? sc1[j] : 0.f) - ss1[i] * sB[e]) * invmu;
    }
    // M2 = mu*I + diag(s2) G0 diag(s2);  b2 = diag(s2) G0 c2
    float g2 = 0.f;
    if (lane < 8) {
#pragma unroll
      for (int i = 0; i < 8; ++i) g2 += sG0[lane * 8 + i] * sc2[i];
    }
    __syncthreads();
    for (int e = lane; e < 64; e += 32) {
      int i = e >> 3, j = e & 7;
      sM[e] = ss2[i] * sG0[e] * ss2[j] + ((i == j) ? mu_ : 0.f);
    }
    if (lane < 8) sb2[lane] = ss2[lane] * g2;
    __syncthreads();
    solve8(sM, sb2, 1, lane);
    if (lane < 8) swcn[lane] = (sc2[lane] - ss2[lane] * sb2[lane]) * invmu;
    __syncthreads();
  };

  // ---------------- outer SCA loop -----------------------------------------
  for (int outer = 0; outer < OUTER; ++outer) {
    matmulA(sW);  // H^T P = G0 W
    if (lane < 8) {
      float Ht = 0.f;
#pragma unroll
      for (int i = 0; i < 8; ++i) Ht += sG0[lane * 8 + i] * swc[i];
      float Tk = 1.0f;
#pragma unroll
      for (int t = 0; t < 8; ++t) {
        float a = sA[lane * 8 + t];
        Tk += a * a;
      }
      float d = sA[lane * 8 + lane];
      float Tck = Tk + Ht * Ht;
      float ak = Tk / (Tk - d * d) - 1.0f;
      float ack = Ht * Ht / Tk;
      float sqak = sqrtf(1.0f + ak);
      float sqack = sqrtf(1.0f + ack);
      float bk = sqak * d / Tk;
      float bck = sqack * Ht / Tck;
      sbk2[lane] = bk * bk;
      sbck2[lane] = bck * bck;
      sc1[lane] = sqak * bk;
      sc2b[lane] = sqack * bck;
      sres[lane] = log2f(1.0f + ack) - ack;
    }
    __syncthreads();

    for (int in = 0; in < INNER; ++in) {
      updP(mu);
      matmulA(sWn);
      if (lane < 8) {
        float Ht = 0.f;
#pragma unroll
        for (int i = 0; i < 8; ++i) Ht += sG0[lane * 8 + i] * swcn[i];
        float Tkn = 1.0f;
#pragma unroll
        for (int t = 0; t < 8; ++t) {
          float a = sA[lane * 8 + t];
          Tkn += a * a;
        }
        float Tckc = Tkn + Ht * Ht;
        sE[lane] = -sbck2[lane] * Tckc + 2.0f * sc2b[lane] * Ht + sres[lane];
        sHt[lane] = Ht;
      }
      float part = 0.f;  // sum(P^2) = sum_ij (G0 Wn)_ij * Wn_ij
      for (int e = lane; e < 64; e += 32) part += sA[e] * sWn[e];
      sred[lane] = part;
      __syncthreads();
      float sumP2 = 0.f;
#pragma unroll
      for (int i = 0; i < 32; ++i) sumP2 += sred[i];
      float sumpc2 = 0.f;  // pc^T pc = wc^T G0 wc
#pragma unroll
      for (int i = 0; i < 8; ++i) sumpc2 += sHt[i] * swcn[i];
      float mun = (sumP2 + sumpc2) / Pt * mu;
      // lambda update (uniform across lanes)
      float E[8];
#pragma unroll
      for (int i = 0; i < 8; ++i) E[i] = sE[i];
      float mE = E[0];
#pragma unroll
      for (int i = 1; i < 8; ++i) mE = fminf(mE, E[i]);
      float rate[8];
#pragma unroll
      for (int i = 0; i < 8; ++i) rate[i] = E[i] - mE + RHO_C + fabsf(mE);
      float y = rate[0];
      int idx = 0;
#pragma unroll
      for (int i = 1; i < 8; ++i)
        if (rate[i] < y) { y = rate[i]; idx = i; }
      float lamn[8];
      float lsum = 0.f;
#pragma unroll
      for (int i = 0; i < 8; ++i) {
        lamn[i] = y / rate[i] * slam[i];
        lsum += lamn[i];
      }
      float lamout = 1.0f - lsum;
      __syncthreads();
      if (lane < 8) slam[lane] = fabsf(lamn[lane] + ((lane == idx) ? lamout : 0.f));
      for (int e = lane; e < 64; e += 32) sW[e] = sWn[e];
      if (lane < 8) swc[lane] = swcn[lane];
      mu = mun;
      __syncthreads();
    }

    // final beamformer update for this outer iteration
    updP(mu);
    for (int e = lane; e < 64; e += 32) sW[e] = sWn[e];
    if (lane < 8) swc[lane] = swcn[lane];
    if (outer == 0) {
      if (lane < 8) slagF[lane] = slam[lane];
      if (lane == 0) slagF[8] = mu;
    }
    if (outer == OUTER - 1) {
      if (lane < 8) slagL[lane] = slam[lane];
      if (lane == 0) slagL[8] = mu;
    }
    __syncthreads();
  }

  // ---------------- count_sum_rate + outputs --------------------------------
  matmulA(sW);
  if (lane < 8) {
    float Ht = 0.f;
#pragma unroll
    for (int i = 0; i < 8; ++i) Ht += sG0[lane * 8 + i] * swc[i];
    float Tk = 1.0f;
#pragma unroll
    for (int t = 0; t < 8; ++t) {
      float a = sA[lane * 8 + t];
      Tk += a * a;
    }
    float Tck = Tk + Ht * Ht;
    float d = sA[lane * 8 + lane];
    sE[lane] = log2f(Tck / Tk);            // rate_c
    sHt[lane] = log2f(Tk / (Tk - d * d));  // rate_p
  }
  __syncthreads();
  float Y = 0.f, rcmin = sE[0];
#pragma unroll
  for (int i = 0; i < 8; ++i) {
    Y += sHt[i];
    rcmin = fminf(rcmin, sE[i]);
  }
  Y += rcmin;
  if (lane == 0) out[b] = Y;

  // PF = H * [W | wc]  (64 x 9), lanes over rows of H
  float* __restrict__ PF = out + B + (size_t)b * NT * (KK + 1);
  for (int t = lane; t < NT; t += 32) {
    float h[8];
#pragma unroll
    for (int i = 0; i < 8; ++i) h[i] = Hb[t * KK + i];
#pragma unroll
    for (int j = 0; j < 8; ++j) {
      float s = 0.f;
#pragma unroll
      for (int i = 0; i < 8; ++i) s += h[i] * sW[i * 8 + j];
      PF[t * (KK + 1) + j] = s;
    }
    float s = 0.f;
#pragma unroll
    for (int i = 0; i < 8; ++i) s += h[i] * swc[i];
    PF[t * (KK + 1) + KK] = s;
  }

  float* __restrict__ Lag =
      out + B + (size_t)B * NT * (KK + 1) + (size_t)b * (KK + 1) * 2;
  if (lane < 9) {
    Lag[lane * 2 + 0] = slagF[lane];
    Lag[lane * 2 + 1] = slagL[lane];
  }
}

extern "C" void kernel_launch(void* const* d_in, const int* in_sizes, int n_in,
                              void* d_out, int out_size, void* d_ws,
                              size_t ws_size, hipStream_t stream) {
  (void)n_in;
  (void)out_size;
  (void)d_ws;
  (void)ws_size;
  const float* H = (const float*)d_in[0];
  const int* Pt = (const int*)d_in[1];
  const int B = in_sizes[0] / (NT * KK);  // 256
  rsma_sca_kernel<<<dim3(B), dim3(32), 0, stream>>>(H, Pt, (float*)d_out, B);
}